// MorphoMLP_32392643346992
// MI455X (gfx1250) — compile-verified
//
#include <hip/hip_runtime.h>

// ---------------------------------------------------------------------------
// MorphoMLP: y = relu(maxplus(relu(maxplus(x, W1)), W2))
//   maxplus(X, W)[m, n] = max_k ( X[m,k] + W[n,k] )   (X:[M,K], W:[N,K] row-major)
// Max-plus has no WMMA mapping (tropical semiring) -> VALU-bound kernel:
//   * CDNA5 async global->LDS staging (ASYNCcnt), double-buffered
//   * padded LDS (stride 36) -> conflict-free ds_load_b128
//   * 4x4 register tiles -> VOPD dual-issue v_dual_add/v_dual_max pairs
//   * k-loop unroll capped at 2 to stay under 256 VGPRs (no s_set_vgpr_msb,
//     multi-wave occupancy per SIMD)
// ---------------------------------------------------------------------------

#if defined(__AMDGCN__) && __has_builtin(__builtin_amdgcn_global_load_async_to_lds_b128)
#define MORPHO_ASYNC 1
#endif

typedef float float4v __attribute__((ext_vector_type(4)));
typedef int   v4i     __attribute__((ext_vector_type(4)));

#define BM 64
#define BN 64
#define KB 32
#define LDS_STRIDE 36              // 32 + 4 pad: conflict-free b128 reads
#define TILE_F (BM * LDS_STRIDE)   // floats per (x or w) tile = 2304

__device__ __forceinline__ void stage16(const float* __restrict__ g, float* l) {
#ifdef MORPHO_ASYNC
  // 16B per lane, global -> LDS, tracked by ASYNCcnt (no VGPR round trip).
  __builtin_amdgcn_global_load_async_to_lds_b128(
      (__attribute__((address_space(1))) v4i*)g,
      (__attribute__((address_space(3))) v4i*)l,
      0, 0);
#else
  *(float4v*)l = *(const float4v*)g;
#endif
}

__device__ __forceinline__ void wait_async() {
#ifdef MORPHO_ASYNC
#if __has_builtin(__builtin_amdgcn_s_wait_asynccnt)
  __builtin_amdgcn_s_wait_asynccnt(0);
#else
  asm volatile("s_wait_asynccnt 0" ::: "memory");
#endif
#endif
}

// Stage one KB-wide chunk of X (BM rows) and W (BN rows) into LDS.
// 256 threads * 4 b128 transfers = (64+64) rows * 32 floats.
__device__ __forceinline__ void stage_chunk(const float* __restrict__ X,
                                            const float* __restrict__ W,
                                            int M0, int N0, int K, int k0,
                                            float* xs, float* ws, int tid) {
#pragma unroll
  for (int r = 0; r < 2; ++r) {
    const int t   = tid + 256 * r;
    const int row = t >> 3;        // 0..63
    const int c4  = t & 7;         // 0..7 float4 column
    stage16(X + (size_t)(M0 + row) * K + k0 + c4 * 4,
            xs + row * LDS_STRIDE + c4 * 4);
    stage16(W + (size_t)(N0 + row) * K + k0 + c4 * 4,
            ws + row * LDS_STRIDE + c4 * 4);
  }
}

__global__ __launch_bounds__(256) void morpho_relu_kernel(
    const float* __restrict__ X,   // [M, K]
    const float* __restrict__ W,   // [N, K]
    float* __restrict__ Out,       // [M, N]
    int M, int N, int K) {
  __shared__ __align__(16) float smem[4 * TILE_F];  // 2 buffers * (xs | ws)

  const int tid = threadIdx.x;
  const int tx  = tid & 15;   // n-tile index
  const int ty  = tid >> 4;   // m-tile index
  const int N0  = blockIdx.x * BN;
  const int M0  = blockIdx.y * BM;

  float acc[4][4];
#pragma unroll
  for (int a = 0; a < 4; ++a)
#pragma unroll
    for (int b = 0; b < 4; ++b) acc[a][b] = -3.402823466e+38f;

  const int nch = K / KB;
  float* buf0 = smem;
  float* buf1 = smem + 2 * TILE_F;

  // Prologue: fill buffer 0
  stage_chunk(X, W, M0, N0, K, 0, buf0, buf0 + TILE_F, tid);
  wait_async();
  __syncthreads();

  for (int c = 0; c < nch; ++c) {
    float* cur = (c & 1) ? buf1 : buf0;
    float* nxt = (c & 1) ? buf0 : buf1;

    // Prefetch next chunk while computing this one (overlap via ASYNCcnt).
    if (c + 1 < nch)
      stage_chunk(X, W, M0, N0, K, (c + 1) * KB, nxt, nxt + TILE_F, tid);

    const float* xs = cur;
    const float* ws = cur + TILE_F;

    // Unroll capped at 2 (8 k-values live) to keep VGPR count < 256:
    // full unroll hoists 64 float4 LDS loads and spills past v255.
#pragma unroll 2
    for (int kk = 0; kk < KB; kk += 4) {
      float4v xr[4], wr[4];
#pragma unroll
      for (int a = 0; a < 4; ++a)
        xr[a] = *(const float4v*)(xs + (ty * 4 + a) * LDS_STRIDE + kk);
#pragma unroll
      for (int b = 0; b < 4; ++b)
        wr[b] = *(const float4v*)(ws + (tx * 4 + b) * LDS_STRIDE + kk);
#pragma unroll
      for (int a = 0; a < 4; ++a)
#pragma unroll
        for (int b = 0; b < 4; ++b) {
          acc[a][b] = fmaxf(acc[a][b], xr[a].x + wr[b].x);
          acc[a][b] = fmaxf(acc[a][b], xr[a].y + wr[b].y);
          acc[a][b] = fmaxf(acc[a][b], xr[a].z + wr[b].z);
          acc[a][b] = fmaxf(acc[a][b], xr[a].w + wr[b].w);
        }
    }

    if (c + 1 < nch) {
      wait_async();        // next chunk landed in LDS
      __syncthreads();     // all waves done reading 'cur', data in 'nxt' visible
    }
  }

  // Epilogue: fused ReLU, vectorized b128 stores.
#pragma unroll
  for (int a = 0; a < 4; ++a) {
    float4v o;
    o.x = fmaxf(acc[a][0], 0.0f);
    o.y = fmaxf(acc[a][1], 0.0f);
    o.z = fmaxf(acc[a][2], 0.0f);
    o.w = fmaxf(acc[a][3], 0.0f);
    *(float4v*)(Out + (size_t)(M0 + ty * 4 + a) * N + N0 + tx * 4) = o;
  }
}

// ---------------------------------------------------------------------------
// Host launcher: two passes of the same kernel; intermediate h in d_ws (2 MB).
// ---------------------------------------------------------------------------
extern "C" void kernel_launch(void* const* d_in, const int* in_sizes, int n_in,
                              void* d_out, int out_size, void* d_ws, size_t ws_size,
                              hipStream_t stream) {
  (void)in_sizes; (void)n_in; (void)out_size; (void)ws_size;

  constexpr int B = 512, IN = 512, HID = 1024, OUT = 512;

  const float* x  = (const float*)d_in[0];   // [B, IN]
  const float* W1 = (const float*)d_in[1];   // [HID, IN]
  const float* W2 = (const float*)d_in[2];   // [OUT, HID]
  float* h = (float*)d_ws;                   // [B, HID] intermediate
  float* y = (float*)d_out;                  // [B, OUT]

  dim3 blk(256);
  morpho_relu_kernel<<<dim3(HID / BN, B / BM), blk, 0, stream>>>(x, W1, h, B, HID, IN);
  morpho_relu_kernel<<<dim3(OUT / BN, B / BM), blk, 0, stream>>>(h, W2, y, B, OUT, HID);
}